// ContrastCELoss_nomem_10376640987314
// MI455X (gfx1250) — compile-verified
//
#include <hip/hip_runtime.h>
#include <math.h>

// ---------------- problem constants ----------------
#define B_   8
#define D_   32
#define N_   65536          // 256*256
#define C_   36
#define K_   50
#define P_   (C_ * K_)      // 1800 anchors per batch
#define MT_  ((P_ + 15) / 16) // 113 row tiles of 16
#define TEMP_      0.1f
#define BASE_TEMP_ 0.07f
#define LOSS_SCALE (TEMP_ / BASE_TEMP_)

// seg-sum WMMA kernel geometry
#define WGS_PB 32                    // workgroups per batch
#define WAVES_ 8                     // waves per workgroup
#define WPX    (N_ / (WGS_PB * WAVES_)) // 256 pixels per wave
#define TPX    32                    // pixels per K-tile

// workspace layout (in 4-byte elements)
#define WS_COUNTS   0                       // int   [B_*C_]          = 288
#define WS_SEGSUM   (WS_COUNTS + B_*C_)     // float [B_*C_*D_]       = 9216
#define WS_SEGSC    (WS_SEGSUM + B_*C_*D_)  // float [B_*C_*D_]       = 9216
#define WS_PIXPOS   (WS_SEGSC + B_*C_*D_)   // int   [B_*C_*K_]       = 14400
#define WS_LOSS     (WS_PIXPOS + B_*C_*K_)  // float [B_]             = 8
#define WS_TOTAL    (WS_LOSS + B_)

typedef __attribute__((ext_vector_type(2))) float v2f;
typedef __attribute__((ext_vector_type(8))) float v8f;

static __device__ __forceinline__ v8f wmma4(v2f a, v2f b, v8f c) {
    return __builtin_amdgcn_wmma_f32_16x16x4_f32(false, a, false, b, (short)0, c,
                                                 false, false);
}

// ---------------- kernel 1: zero workspace ----------------
__global__ void k_zero(float* ws, int n) {
    int i = blockIdx.x * blockDim.x + threadIdx.x;
    if (i < n) ws[i] = 0.0f;
}

// ---------------- kernel 2a: class counts (labels only, 2 MiB traffic) ----------------
__global__ void k_counts(const int* __restrict__ labels, int* __restrict__ counts) {
    __shared__ int cntl[C_];
    const int tid = threadIdx.x;
    const int b     = blockIdx.x >> 4;
    const int chunk = blockIdx.x & 15;
    const int base  = chunk * 4096;
    for (int i = tid; i < C_; i += 256) cntl[i] = 0;
    __syncthreads();
    for (int i = 0; i < 16; ++i)
        atomicAdd(&cntl[labels[b * N_ + base + i * 256 + tid]], 1);
    __syncthreads();
    for (int i = tid; i < C_; i += 256) atomicAdd(&counts[b * C_ + i], cntl[i]);
}

// ---------------- kernel 2b: seg sums as WMMA GEMM: onehot^T (48xN) x feats (Nx32) ----
// grid: B_*WGS_PB workgroups of 256 threads (8 waves); each wave owns WPX pixels.
__global__ __launch_bounds__(256) void k_segsum_wmma(const float* __restrict__ emb,
                                                     const int*   __restrict__ labels,
                                                     float* __restrict__ seg_sum) {
    __shared__ __align__(16) float sE[WAVES_][TPX * 32]; // per-wave 32d x 32n tile (32 KB)
    __shared__ int   sLab[WAVES_][TPX];
    __shared__ float sSeg[48 * 32];                      // WG partial (6 KB)

    const int tid  = threadIdx.x;
    const int lane = tid & 31;
    const int w    = tid >> 5;
    const int b    = blockIdx.x / WGS_PB;
    const int wgc  = blockIdx.x % WGS_PB;
    const int n_base = (wgc * WAVES_ + w) * WPX;

    for (int i = tid; i < 48 * 32; i += 256) sSeg[i] = 0.0f;
    __syncthreads();

    const int half = lane >> 4;   // K-offset select (A/B frag layout)
    const int r16  = lane & 15;   // row / col within 16-tile

    v8f acc[3][2] = {};           // [class tile 0..2][d tile 0..1]

    for (int t0 = 0; t0 < WPX; t0 += TPX) {
        const int n0 = n_base + t0;

        // stage labels + 32x32 embedding tile (coalesced b128 loads)
        sLab[w][lane] = labels[b * N_ + n0 + lane];
#pragma unroll
        for (int it = 0; it < 8; ++it) {
            const int flat = it * 128 + lane * 4;
            const int d = flat >> 5, noff = flat & 31;
            const float4 v = *(const float4*)(emb + (size_t)(b * D_ + d) * N_ + n0 + noff);
            *(float4*)(&sE[w][d * 32 + noff]) = v;
        }
        // speculative prefetch of next tile (gfx1250 global_prefetch_b8)
        __builtin_prefetch(emb + (size_t)(b * D_) * N_ + n0 + TPX, 0, 1);
        __syncthreads();

        // 8 K-steps of 4 pixels: A = onehot (class x pixel), B = feats (pixel x d)
#pragma unroll
        for (int kk = 0; kk < 8; ++kk) {
            const int kbase = kk * 4 + 2 * half; // lanes 0-15: K=k0,k0+1 ; 16-31: K=k0+2,k0+3
            const int l0 = sLab[w][kbase];
            const int l1 = sLab[w][kbase + 1];
            v2f bf0 = { sE[w][(0 * 16 + r16) * 32 + kbase], sE[w][(0 * 16 + r16) * 32 + kbase + 1] };
            v2f bf1 = { sE[w][(1 * 16 + r16) * 32 + kbase], sE[w][(1 * 16 + r16) * 32 + kbase + 1] };
#pragma unroll
            for (int m = 0; m < 3; ++m) {
                const int c = m * 16 + r16;
                v2f a = { (l0 == c) ? 1.0f : 0.0f, (l1 == c) ? 1.0f : 0.0f };
                acc[m][0] = wmma4(a, bf0, acc[m][0]);
                acc[m][1] = wmma4(a, bf1, acc[m][1]);
            }
        }
        __syncthreads();
    }

    // flush wave partials into WG LDS accumulator
    // C/D layout: VGPR r, lane L -> class = m*16 + r + 8*(L>>4), d = t*16 + (L&15)
#pragma unroll
    for (int m = 0; m < 3; ++m)
#pragma unroll
        for (int t = 0; t < 2; ++t)
#pragma unroll
            for (int r = 0; r < 8; ++r) {
                const int c = m * 16 + r + 8 * half;
                if (c < C_) atomicAdd(&sSeg[c * 32 + (t * 16 + r16)], acc[m][t][r]);
            }
    __syncthreads();

    for (int i = tid; i < C_ * D_; i += 256)
        atomicAdd(&seg_sum[b * C_ * D_ + i], sSeg[i]);
}

// ---------------- kernel 3: first-K position selection ----------------
__global__ void k_select(const int* __restrict__ labels,
                         int* __restrict__ pix_pos) {
    const int wg   = blockIdx.x;
    const int b    = wg / C_;
    const int c    = wg % C_;
    const int lane = threadIdx.x;

    int base = 0;
    for (int start = 0; start < N_ && base < K_; start += 32) {
        const int  lbl   = labels[b * N_ + start + lane];
        const bool match = (lbl == c);
        const unsigned mask   = __builtin_amdgcn_ballot_w32(match);
        const int      prefix = __builtin_popcount(mask & ((1u << lane) - 1u));
        if (match) {
            const int slot = base + prefix;
            if (slot < K_) pix_pos[(b * C_ + c) * K_ + slot] = start + lane;
        }
        base += __builtin_popcount(mask);
    }
}

// ---------------- kernel 4: normalize seg means, fold in 1/TEMP ----------------
__global__ void k_normalize(const float* __restrict__ seg_sum,
                            const int*   __restrict__ counts,
                            float* __restrict__ seg_sc) {
    const int i = blockIdx.x * blockDim.x + threadIdx.x;
    if (i < B_ * C_ * D_) {
        const int   bc  = i / D_;
        const float cnt = fmaxf((float)counts[bc], 1.0f);
        seg_sc[i] = seg_sum[i] / (cnt * TEMP_);
    }
}

// ---------------- kernel 5: WMMA logits + masked log-softmax loss ----------------
__global__ void k_logits_loss(const float* __restrict__ emb,
                              const int*   __restrict__ counts,
                              const float* __restrict__ seg_sc,
                              const int*   __restrict__ pix_pos,
                              float* __restrict__ loss_sum) {
    __shared__ float sA[16 * 32];   // [row][d]
    __shared__ float sB[32 * 48];   // [d][class_padded]
    __shared__ float sL[16 * 48];   // logits tile
    __shared__ int   rowPos[16];
    __shared__ int   rowValid[16];
    __shared__ int   cntl[C_];

    const int lane = threadIdx.x;
    const int b    = blockIdx.x / MT_;
    const int mt   = blockIdx.x % MT_;

    for (int i = lane; i < C_; i += 32) cntl[i] = counts[b * C_ + i];
    for (int i = lane; i < 32 * 48; i += 32) sB[i] = 0.0f;
    __syncthreads();

    for (int i = lane; i < C_ * D_; i += 32) {
        const int c = i / D_, d = i % D_;
        sB[d * 48 + c] = seg_sc[(b * C_ + c) * D_ + d];
    }
    if (lane < 16) {
        const int p = mt * 16 + lane;
        int v = 0, pos = 0;
        if (p < P_) {
            const int c = p / K_, k = p % K_;
            if (k < cntl[c]) { v = 1; pos = pix_pos[(b * C_ + c) * K_ + k]; }
        }
        rowValid[lane] = v;
        rowPos[lane]   = pos;
    }
    __syncthreads();

    for (int i = lane; i < 16 * 32; i += 32) {
        const int m = i >> 5, d = i & 31;
        sA[i] = rowValid[m] ? emb[(size_t)(b * D_ + d) * N_ + rowPos[m]] : 0.0f;
    }
    __syncthreads();

    const int row  = lane & 15;
    const int half = lane >> 4;
    v8f acc0 = {}, acc1 = {}, acc2 = {};
#pragma unroll
    for (int kk = 0; kk < 8; ++kk) {
        const int k0 = kk * 4 + 2 * half;
        v2f a  = { sA[row * 32 + k0],       sA[row * 32 + k0 + 1] };
        v2f b0 = { sB[k0 * 48 + row],       sB[(k0 + 1) * 48 + row] };
        v2f b1 = { sB[k0 * 48 + 16 + row],  sB[(k0 + 1) * 48 + 16 + row] };
        v2f b2 = { sB[k0 * 48 + 32 + row],  sB[(k0 + 1) * 48 + 32 + row] };
        acc0 = wmma4(a, b0, acc0);
        acc1 = wmma4(a, b1, acc1);
        acc2 = wmma4(a, b2, acc2);
    }

#pragma unroll
    for (int r = 0; r < 8; ++r) {
        const int M = r + 8 * half;
        sL[M * 48 + row]      = acc0[r];
        sL[M * 48 + 16 + row] = acc1[r];
        sL[M * 48 + 32 + row] = acc2[r];
    }
    __syncthreads();

    if (lane < 16) {
        const int p = mt * 16 + lane;
        if (p < P_) {
            const int c = p / K_, k = p % K_;
            if (k < cntl[c]) {
                float mx = -INFINITY;
                for (int c2 = 0; c2 < C_; ++c2)
                    if (cntl[c2] > 0) mx = fmaxf(mx, sL[lane * 48 + c2]);
                float S = 0.0f;
                for (int c2 = 0; c2 < C_; ++c2)
                    if (cntl[c2] > 0) S += expf(sL[lane * 48 + c2] - mx);
                const float lp = (sL[lane * 48 + c] - mx) - logf(S);
                atomicAdd(&loss_sum[b], -LOSS_SCALE * lp);
            }
        }
    }
}

// ---------------- kernel 6: finalize ----------------
__global__ void k_finalize(const int* __restrict__ counts,
                           const float* __restrict__ loss_sum,
                           float* __restrict__ out) {
    if (threadIdx.x == 0 && blockIdx.x == 0) {
        float acc = 0.0f;
        for (int b = 0; b < B_; ++b) {
            int nv = 0;
            for (int c = 0; c < C_; ++c) {
                const int cnt = counts[b * C_ + c];
                nv += (cnt < K_) ? cnt : K_;
            }
            acc += loss_sum[b] / fmaxf((float)nv, 1.0f);
        }
        out[0] = acc / (float)B_;
    }
}

// ---------------- launcher ----------------
extern "C" void kernel_launch(void* const* d_in, const int* in_sizes, int n_in,
                              void* d_out, int out_size, void* d_ws, size_t ws_size,
                              hipStream_t stream) {
    const float* emb    = (const float*)d_in[0];   // (B, D, H, W) f32
    const int*   labels = (const int*)d_in[1];     // (B, H, W)    i32

    float* wsf      = (float*)d_ws;
    int*   counts   = (int*)  (wsf + WS_COUNTS);
    float* seg_sum  =          wsf + WS_SEGSUM;
    float* seg_sc   =          wsf + WS_SEGSC;
    int*   pix_pos  = (int*)  (wsf + WS_PIXPOS);
    float* loss_sum =          wsf + WS_LOSS;

    k_zero<<<(WS_TOTAL + 255) / 256, 256, 0, stream>>>(wsf, WS_TOTAL);
    k_counts<<<B_ * 16, 256, 0, stream>>>(labels, counts);
    k_segsum_wmma<<<B_ * WGS_PB, 256, 0, stream>>>(emb, labels, seg_sum);
    k_select<<<B_ * C_, 32, 0, stream>>>(labels, pix_pos);
    k_normalize<<<(B_ * C_ * D_ + 255) / 256, 256, 0, stream>>>(seg_sum, counts, seg_sc);
    k_logits_loss<<<B_ * MT_, 32, 0, stream>>>(emb, counts, seg_sc, pix_pos, loss_sum);
    k_finalize<<<1, 1, 0, stream>>>(counts, loss_sum, (float*)d_out);
}